// GAT_22454089024165
// MI455X (gfx1250) — compile-verified
//
#include <hip/hip_runtime.h>
#include <hip/hip_bf16.h>
#include <math.h>

// GAT on star-graph + self-loop adjacency: the dense softmax collapses to a
// 2-way softmax per row (i>0) plus one global softmax reduction for row 0.
// Dot products x·a are batched through V_WMMA_F32_16X16X4_F32.

#define GAT_D 64

typedef __attribute__((ext_vector_type(2))) float v2f;
typedef __attribute__((ext_vector_type(8))) float v8f;

// ---------------------------------------------------------------------------
// Kernel 1: per-row dot products via WMMA f32 16x16x4.
// One wave handles 16 rows. B-matrix column 0 = a_hi (=> h1), column 1 = a_lo
// (=> h2), columns 2..15 zero. 16 K-steps of 4 cover D=64.
// Fragment layouts per CDNA5 ISA 7.12.2:
//   A 16x4 : lane m (0-15) rows, half=lane>>4 selects K offset {0,2};
//            VGPR0 = K+0, VGPR1 = K+1.
//   B 4x16 : VGPR v holds row K=v (lanes 0-15) and K=v+2 (lanes 16-31),
//            column = lane&15  (mirrors the A layout).
//   D 16x16: lane&15 = column N, VGPR v = row v (+8 for lanes 16-31).
// B-fragment build is fully branchless: base pointer + lane mask selected
// once before the loop, unconditional b64 loads inside (EXEC stays all-1s,
// no per-iteration saveexec predication).
// ---------------------------------------------------------------------------
__global__ __launch_bounds__(256) void gat_dots_wmma(
    const float* __restrict__ x,   // [n, 64]
    const float* __restrict__ A,   // [128]  (A[:64]=a_lo, A[64:]=a_hi)
    float* __restrict__ g1,        // [n]  x . a_hi
    float* __restrict__ g2) {      // [n]  x . a_lo
  const int lane = threadIdx.x & 31;
  const int wave = threadIdx.x >> 5;
  const int base = (blockIdx.x * 8 + wave) * 16;   // first row of this tile
  const int half = lane >> 4;                      // 0 or 1
  const int m    = lane & 15;

  const float* arow = x + (base + m) * GAT_D;
  const float* a_lo = A;
  const float* a_hi = A + GAT_D;

  // Lane's B source: column 0 reads a_hi, column 1 reads a_lo, others are
  // forced to zero via the multiply mask (loads stay unconditional).
  const float* bptr  = (m == 0) ? a_hi : a_lo;
  const float  bmask = (m < 2) ? 1.0f : 0.0f;

  v8f acc = {};
#pragma unroll
  for (int k0 = 0; k0 < GAT_D; k0 += 4) {
    const int kk = k0 + half * 2;          // even -> 8B aligned
    const v2f afrag = *(const v2f*)(arow + kk);
    v2f bfrag = *(const v2f*)(bptr + kk);
    bfrag.x *= bmask;
    bfrag.y *= bmask;
    acc = __builtin_amdgcn_wmma_f32_16x16x4_f32(
        /*neg_a=*/false, afrag, /*neg_b=*/false, bfrag,
        /*c_mod=*/(short)0, acc, /*reuse_a=*/false, /*reuse_b=*/false);
  }

  // Columns 0 (a_hi -> g1) and 1 (a_lo -> g2) live in lanes {0,16} / {1,17}.
  if (m < 2) {
    float* dst = (m == 0) ? g1 : g2;
    const int r0 = base + half * 8;
#pragma unroll
    for (int v = 0; v < 8; ++v) dst[r0 + v] = acc[v];
  }
}

// ---------------------------------------------------------------------------
// Kernel 2: row-0 output. att(0,:) = softmax(score) over all n nodes;
// out0[c] = sum_j softmax(score)_j * h[j][c]. Single block, fixed-order
// (deterministic) tree reductions.
// ---------------------------------------------------------------------------
__global__ __launch_bounds__(1024) void gat_row0(
    const float* __restrict__ h,      // [n, 64]
    const float* __restrict__ score,  // [n]   (= h2 of this layer)
    float* __restrict__ out0,         // [64]  row 0 of the layer output
    int n) {
  __shared__ float sm[1024];
  const int tid = threadIdx.x;

  // pass 1: global max
  float mx = -3.4e38f;
  for (int j = tid; j < n; j += 1024) mx = fmaxf(mx, score[j]);
  sm[tid] = mx;
  __syncthreads();
  for (int s = 512; s > 0; s >>= 1) {
    if (tid < s) sm[tid] = fmaxf(sm[tid], sm[tid + s]);
    __syncthreads();
  }
  const float M = sm[0];
  __syncthreads();

  // pass 2: sum of exp
  float ssum = 0.0f;
  for (int j = tid; j < n; j += 1024) ssum += expf(score[j] - M);
  sm[tid] = ssum;
  __syncthreads();
  for (int s = 512; s > 0; s >>= 1) {
    if (tid < s) sm[tid] += sm[tid + s];
    __syncthreads();
  }
  const float S = sm[0];
  __syncthreads();

  // pass 3: weighted column sums. thread (ty,tx): column tx, rows ty+16k.
  const int tx = tid & 63;
  const int ty = tid >> 6;  // 0..15
  float acc = 0.0f;
  for (int j = ty; j < n; j += 16) acc += expf(score[j] - M) * h[j * GAT_D + tx];
  sm[ty * 64 + tx] = acc;
  __syncthreads();
  for (int s = 8; s > 0; s >>= 1) {
    if (ty < s) sm[ty * 64 + tx] += sm[(ty + s) * 64 + tx];
    __syncthreads();
  }
  if (ty == 0) out0[tx] = sm[tx] / S;
}

// ---------------------------------------------------------------------------
// Kernel 3: rows i>0. Attention reduces to a 2-way softmax over
// a = g1[i]+g2[0] (edge to node 0) and b = g1[i]+g2[i] (self loop):
// out[i] = w0*h[0] + wi*h[i]. One wave per row, float2 per lane (coalesced).
// ---------------------------------------------------------------------------
__global__ __launch_bounds__(256) void gat_rows(
    const float* __restrict__ h,   // [n, 64]
    const float* __restrict__ g1,  // [n]
    const float* __restrict__ g2,  // [n]
    float* __restrict__ out,       // [n, 64] (row 0 written elsewhere)
    int n) {
  const int lane = threadIdx.x & 31;
  const int wave = threadIdx.x >> 5;
  const int r = blockIdx.x * 8 + wave;
  if (r == 0 || r >= n) return;

  const float a = g1[r] + g2[0];
  const float b = g1[r] + g2[r];
  const float mm = fmaxf(a, b);
  float w0 = expf(a - mm);
  float wi = expf(b - mm);
  const float inv = 1.0f / (w0 + wi);
  w0 *= inv;
  wi *= inv;

  const int c = lane * 2;
  const float2 x0 = *(const float2*)(h + c);
  const float2 xr = *(const float2*)(h + r * GAT_D + c);
  float2 y;
  y.x = w0 * x0.x + wi * xr.x;
  y.y = w0 * x0.y + wi * xr.y;
  *(float2*)(out + r * GAT_D + c) = y;
}

extern "C" void kernel_launch(void* const* d_in, const int* in_sizes, int n_in,
                              void* d_out, int out_size, void* d_ws, size_t ws_size,
                              hipStream_t stream) {
  const float* x  = (const float*)d_in[0];  // [n,64]
  const float* A1 = (const float*)d_in[1];  // [128]
  const float* A2 = (const float*)d_in[2];  // [128]
  float* out = (float*)d_out;               // [n,64]

  const int n = in_sizes[0] / GAT_D;        // 8192

  // workspace: x1 [n*64] | g1 [n] | g2 [n] | p1 [n] | p2 [n]  (~2.2 MB)
  float* w  = (float*)d_ws;
  float* x1 = w;
  float* g1 = x1 + (size_t)n * GAT_D;
  float* g2 = g1 + n;
  float* p1 = g2 + n;
  float* p2 = p1 + n;

  const int dots_blocks = n / 128;  // 8 waves/block * 16 rows/wave
  const int rows_blocks = (n + 7) / 8;

  // ---- layer 1: x -> x1 ----
  gat_dots_wmma<<<dots_blocks, 256, 0, stream>>>(x, A1, g1, g2);
  gat_row0<<<1, 1024, 0, stream>>>(x, g2, x1, n);          // x1 row 0
  gat_rows<<<rows_blocks, 256, 0, stream>>>(x, g1, g2, x1, n);

  // ---- layer 2: x1 -> out ----
  gat_dots_wmma<<<dots_blocks, 256, 0, stream>>>(x1, A2, p1, p2);
  gat_row0<<<1, 1024, 0, stream>>>(x1, p2, out, n);        // out row 0
  gat_rows<<<rows_blocks, 256, 0, stream>>>(x1, p1, p2, out, n);
}